// TopKDiceLoss_24893630447856
// MI455X (gfx1250) — compile-verified
//
#include <hip/hip_runtime.h>
#include <cstdint>

#define BATCH     32
#define HW        512
#define NPIX      (HW*HW)          // 262144 pixels per batch
#define CHUNKS    64
#define CHUNK_PIX (NPIX / CHUNKS)  // 4096
#define TILE_PIX  1024
#define TILES     (CHUNK_PIX / TILE_PIX) // 4
#define THREADS   256
#define SENTINEL  0xFFFFFFFFu

typedef float v2f __attribute__((ext_vector_type(2)));
typedef float v8f __attribute__((ext_vector_type(8)));

// ---------------------------------------------------------------------------
// P1: stream logits+target via async global->LDS DMA (double buffered),
// compute p = sigmoid(l1-l0), accumulate per-batch sums + level-1 histogram,
// store p bit patterns (sentinel for background) to workspace.
// ---------------------------------------------------------------------------
__global__ __launch_bounds__(THREADS)
void pass1_stream(const float* __restrict__ logits, const int* __restrict__ target,
                  unsigned* __restrict__ pbits,
                  float* __restrict__ sumAll, float* __restrict__ sumFg,
                  unsigned* __restrict__ nFg, unsigned* __restrict__ hist1)
{
    __shared__ __align__(16) float    l0t[2][TILE_PIX];
    __shared__ __align__(16) float    l1t[2][TILE_PIX];
    __shared__ __align__(16) unsigned tgt[2][TILE_PIX];
    __shared__ unsigned hist[2048];
    __shared__ float    redf[2][8];
    __shared__ unsigned redu[8];

    const int b   = blockIdx.y;
    const int tid = threadIdx.x;

    for (int i = tid; i < 2048; i += THREADS) hist[i] = 0;
    __syncthreads();

    const unsigned pix0 = blockIdx.x * CHUNK_PIX;
    const unsigned long long l0base =
        (unsigned long long)(uintptr_t)(logits + (size_t)b * 2 * NPIX);
    const unsigned long long l1base = l0base + (unsigned long long)NPIX * 4ull;
    const unsigned long long tbase  =
        (unsigned long long)(uintptr_t)(target + (size_t)b * NPIX);
    unsigned* __restrict__ outp = pbits + (size_t)b * NPIX;

    float sp = 0.0f, sf = 0.0f;
    unsigned nf = 0;

    auto issue = [&](int tile, int buf) {
        unsigned boff = (pix0 + (unsigned)tile * TILE_PIX + (unsigned)tid * 4u) * 4u;
        unsigned a0 = (unsigned)(uintptr_t)&l0t[buf][tid * 4];
        unsigned a1 = (unsigned)(uintptr_t)&l1t[buf][tid * 4];
        unsigned a2 = (unsigned)(uintptr_t)&tgt[buf][tid * 4];
        asm volatile("global_load_async_to_lds_b128 %0, %1, %2"
                     :: "v"(a0), "v"(boff), "s"(l0base) : "memory");
        asm volatile("global_load_async_to_lds_b128 %0, %1, %2"
                     :: "v"(a1), "v"(boff), "s"(l1base) : "memory");
        asm volatile("global_load_async_to_lds_b128 %0, %1, %2"
                     :: "v"(a2), "v"(boff), "s"(tbase) : "memory");
    };

    issue(0, 0);
    for (int t = 0; t < TILES; ++t) {
        const int buf = t & 1;
        if (t + 1 < TILES) {
            // buffer we are about to overwrite was fully consumed (ds reads done)
            asm volatile("s_wait_dscnt 0x0" ::: "memory");
            issue(t + 1, buf ^ 1);
            // previous 3 async ops must be complete; 3 newly issued may remain
            asm volatile("s_wait_asynccnt 0x3" ::: "memory");
        } else {
            asm volatile("s_wait_asynccnt 0x0" ::: "memory");
        }

        float4 A0 = *(const float4*)&l0t[buf][tid * 4];
        float4 A1 = *(const float4*)&l1t[buf][tid * 4];
        uint4  Tg = *(const uint4*)&tgt[buf][tid * 4];

        float d0[4] = {A1.x - A0.x, A1.y - A0.y, A1.z - A0.z, A1.w - A0.w};
        unsigned tv[4] = {Tg.x, Tg.y, Tg.z, Tg.w};
        uint4 ob;
        unsigned* obp = (unsigned*)&ob;
#pragma unroll
        for (int j = 0; j < 4; ++j) {
            float p = 1.0f / (1.0f + __expf(-d0[j]));  // softmax(axis=1)[:,1]
            sp += p;
            unsigned pb = SENTINEL;
            if (tv[j] == 1u) {
                sf += p;
                nf++;
                unsigned bits = __float_as_uint(p);    // p in (0,1): bits monotone
                atomicAdd(&hist[bits >> 20], 1u);      // bits[30:20] (bit31 == 0)
                pb = bits;
            }
            obp[j] = pb;
        }
        *(uint4*)&outp[pix0 + (unsigned)t * TILE_PIX + (unsigned)tid * 4u] = ob;
    }

    // wave32 reduction, then cross-wave via LDS
    for (int o = 16; o > 0; o >>= 1) {
        sp += __shfl_xor(sp, o);
        sf += __shfl_xor(sf, o);
        nf += __shfl_xor(nf, o);
    }
    const int lane = tid & 31, wid = tid >> 5;
    if (lane == 0) { redf[0][wid] = sp; redf[1][wid] = sf; redu[wid] = nf; }
    __syncthreads();
    if (tid == 0) {
        float s0 = 0.0f, s1 = 0.0f; unsigned s2 = 0;
        for (int w = 0; w < 8; ++w) { s0 += redf[0][w]; s1 += redf[1][w]; s2 += redu[w]; }
        atomicAdd(&sumAll[b], s0);
        atomicAdd(&sumFg[b], s1);
        atomicAdd(&nFg[b], s2);
    }
    __syncthreads();
    for (int i = tid; i < 2048; i += THREADS) {
        unsigned h = hist[i];
        if (h) atomicAdd(&hist1[b * 2048 + i], h);
    }
}

// ---------------------------------------------------------------------------
// Scans: locate bin containing the k-th smallest foreground p (radix select).
// ---------------------------------------------------------------------------
__global__ void scan1_k(const unsigned* __restrict__ hist1, const unsigned* __restrict__ nFg,
                        unsigned* __restrict__ sel1, unsigned* __restrict__ rank2)
{
    const int b = threadIdx.x;
    if (b >= BATCH) return;
    const unsigned n = nFg[b];
    if (n == 0) { sel1[b] = SENTINEL; rank2[b] = 0; return; }
    int k = (int)floorf((float)n * 0.1f);
    if (k < 1) k = 1;
    unsigned cum = 0;
    for (int i = 0; i < 2048; ++i) {
        unsigned c = hist1[b * 2048 + i];
        if (cum + c >= (unsigned)k) { sel1[b] = (unsigned)i; rank2[b] = (unsigned)k - cum; return; }
        cum += c;
    }
    sel1[b] = 2047u; rank2[b] = 1u;  // unreachable fallback
}

__global__ void scan2_k(const unsigned* __restrict__ hist2, const unsigned* __restrict__ sel1,
                        const unsigned* __restrict__ rank2,
                        unsigned* __restrict__ sel2, unsigned* __restrict__ rank3)
{
    const int b = threadIdx.x;
    if (b >= BATCH) return;
    if (sel1[b] == SENTINEL) { sel2[b] = 0; rank3[b] = 0; return; }
    const unsigned k = rank2[b];
    unsigned cum = 0;
    for (int i = 0; i < 2048; ++i) {
        unsigned c = hist2[b * 2048 + i];
        if (cum + c >= k) { sel2[b] = (unsigned)i; rank3[b] = k - cum; return; }
        cum += c;
    }
    sel2[b] = 2047u; rank3[b] = 1u;
}

__global__ void scan3_k(const unsigned* __restrict__ hist3, const unsigned* __restrict__ sel1,
                        const unsigned* __restrict__ sel2, const unsigned* __restrict__ rank3,
                        unsigned* __restrict__ thr)
{
    const int b = threadIdx.x;
    if (b >= BATCH) return;
    if (sel1[b] == SENTINEL) { thr[b] = 0u; return; }  // no fg: keep nothing
    const unsigned k = rank3[b];
    unsigned cum = 0;
    for (int i = 0; i < 512; ++i) {
        unsigned c = hist3[b * 512 + i];
        if (cum + c >= k) {
            thr[b] = (sel1[b] << 20) | (sel2[b] << 9) | (unsigned)i;  // exact k-th value
            return;
        }
        cum += c;
    }
    thr[b] = (sel1[b] << 20) | (sel2[b] << 9) | 511u;
}

// ---------------------------------------------------------------------------
// Refinement histograms over stored p bit patterns.
// ---------------------------------------------------------------------------
__global__ __launch_bounds__(THREADS)
void hist2_k(const unsigned* __restrict__ pbits, const unsigned* __restrict__ sel1,
             unsigned* __restrict__ hist2)
{
    __shared__ unsigned hist[2048];
    const int b = blockIdx.y;
    const unsigned s1 = sel1[b];
    if (s1 == SENTINEL) return;
    const int tid = threadIdx.x;
    for (int i = tid; i < 2048; i += THREADS) hist[i] = 0;
    __syncthreads();
    const size_t base = (size_t)b * NPIX + (size_t)blockIdx.x * CHUNK_PIX;
    for (int i = tid * 4; i < CHUNK_PIX; i += THREADS * 4) {
        uint4 v = *(const uint4*)&pbits[base + i];
        unsigned a[4] = {v.x, v.y, v.z, v.w};
#pragma unroll
        for (int j = 0; j < 4; ++j)
            if ((a[j] >> 20) == s1) atomicAdd(&hist[(a[j] >> 9) & 0x7FFu], 1u);
    }
    __syncthreads();
    for (int i = tid; i < 2048; i += THREADS) {
        unsigned h = hist[i];
        if (h) atomicAdd(&hist2[b * 2048 + i], h);
    }
}

__global__ __launch_bounds__(THREADS)
void hist3_k(const unsigned* __restrict__ pbits, const unsigned* __restrict__ sel1,
             const unsigned* __restrict__ sel2, unsigned* __restrict__ hist3)
{
    __shared__ unsigned hist[512];
    const int b = blockIdx.y;
    const unsigned s1 = sel1[b];
    if (s1 == SENTINEL) return;
    const unsigned prefix = (s1 << 11) | sel2[b];   // bits[30:9]
    const int tid = threadIdx.x;
    for (int i = tid; i < 512; i += THREADS) hist[i] = 0;
    __syncthreads();
    const size_t base = (size_t)b * NPIX + (size_t)blockIdx.x * CHUNK_PIX;
    for (int i = tid * 4; i < CHUNK_PIX; i += THREADS * 4) {
        uint4 v = *(const uint4*)&pbits[base + i];
        unsigned a[4] = {v.x, v.y, v.z, v.w};
#pragma unroll
        for (int j = 0; j < 4; ++j)
            if ((a[j] >> 9) == prefix) atomicAdd(&hist[a[j] & 0x1FFu], 1u);
    }
    __syncthreads();
    for (int i = tid; i < 512; i += THREADS) {
        unsigned h = hist[i];
        if (h) atomicAdd(&hist3[b * 512 + i], h);
    }
}

// ---------------------------------------------------------------------------
// P7: sum/count of kept foreground pixels (bits <= thr; ties kept == p > thr dropped)
// ---------------------------------------------------------------------------
__global__ __launch_bounds__(THREADS)
void kept_k(const unsigned* __restrict__ pbits, const unsigned* __restrict__ thr,
            float* __restrict__ keptSum, unsigned* __restrict__ keptCnt)
{
    __shared__ float    redf[8];
    __shared__ unsigned redu[8];
    const int b = blockIdx.y;
    const unsigned T = thr[b];
    const int tid = threadIdx.x;
    const size_t base = (size_t)b * NPIX + (size_t)blockIdx.x * CHUNK_PIX;
    float ks = 0.0f; unsigned kc = 0;
    for (int i = tid * 4; i < CHUNK_PIX; i += THREADS * 4) {
        uint4 v = *(const uint4*)&pbits[base + i];
        unsigned a[4] = {v.x, v.y, v.z, v.w};
#pragma unroll
        for (int j = 0; j < 4; ++j)
            if (a[j] <= T) { ks += __uint_as_float(a[j]); kc++; }  // sentinel > any thr
    }
    for (int o = 16; o > 0; o >>= 1) { ks += __shfl_xor(ks, o); kc += __shfl_xor(kc, o); }
    const int lane = tid & 31, wid = tid >> 5;
    if (lane == 0) { redf[wid] = ks; redu[wid] = kc; }
    __syncthreads();
    if (tid == 0) {
        float s0 = 0.0f; unsigned s1 = 0;
        for (int w = 0; w < 8; ++w) { s0 += redf[w]; s1 += redu[w]; }
        atomicAdd(&keptSum[b], s0);
        atomicAdd(&keptCnt[b], s1);
    }
}

// ---------------------------------------------------------------------------
// P8: per-batch dice, batch mean via V_WMMA_F32_16X16X4_F32 (A=ones, B=dice)
// ---------------------------------------------------------------------------
__global__ void finalize_k(const float* __restrict__ sumAll, const float* __restrict__ sumFg,
                           const float* __restrict__ keptSum, const unsigned* __restrict__ keptCnt,
                           float* __restrict__ out)
{
    const int b = threadIdx.x;  // exactly one wave32, EXEC all ones (WMMA requirement)
    const float inter = keptSum[b];
    const float spk = sumAll[b] - (sumFg[b] - keptSum[b]);  // sum(p*mask)
    const float stk = (float)keptCnt[b];                    // sum(t*mask)
    const float dice = (2.0f * inter + 1e-5f) / (spk + stk + 1e-5f);

    // D = Ones(16x4) x B(4x16): every lane's D[0] = dice[lane%16] + dice[lane%16+16]
    v2f a;  a[0] = 1.0f; a[1] = 1.0f;
    v2f bm; bm[0] = dice; bm[1] = 0.0f;
    v8f c = {};
    c = __builtin_amdgcn_wmma_f32_16x16x4_f32(false, a, false, bm,
                                              (short)0, c, false, false);
    float s = c[0];
    for (int o = 8; o > 0; o >>= 1) s += __shfl_xor(s, o);  // sum 16 column sums
    if (b == 0) out[0] = 1.0f - s * (1.0f / 32.0f);
}

// ---------------------------------------------------------------------------
extern "C" void kernel_launch(void* const* d_in, const int* in_sizes, int n_in,
                              void* d_out, int out_size, void* d_ws, size_t ws_size,
                              hipStream_t stream)
{
    (void)in_sizes; (void)n_in; (void)out_size; (void)ws_size;
    const float* logits = (const float*)d_in[0];
    const int*   target = (const int*)d_in[1];
    float* out = (float*)d_out;

    char* ws = (char*)d_ws;
    unsigned* pbits = (unsigned*)ws;                       // 32 MiB
    char* ctrl = ws + (size_t)BATCH * NPIX * 4;
    float*    sumAll  = (float*)ctrl;
    float*    sumFg   = sumAll  + BATCH;
    float*    keptSum = sumFg   + BATCH;
    unsigned* nFg     = (unsigned*)(keptSum + BATCH);
    unsigned* keptCnt = nFg     + BATCH;
    unsigned* sel1    = keptCnt + BATCH;
    unsigned* rank2   = sel1    + BATCH;
    unsigned* sel2    = rank2   + BATCH;
    unsigned* rank3   = sel2    + BATCH;
    unsigned* thr     = rank3   + BATCH;
    unsigned* hist1   = thr     + BATCH;
    unsigned* hist2   = hist1   + BATCH * 2048;
    unsigned* hist3   = hist2   + BATCH * 2048;
    const size_t ctrlBytes = (size_t)((char*)(hist3 + BATCH * 512) - ctrl);

    hipMemsetAsync(ctrl, 0, ctrlBytes, stream);

    dim3 grid(CHUNKS, BATCH);
    pass1_stream<<<grid, THREADS, 0, stream>>>(logits, target, pbits,
                                               sumAll, sumFg, nFg, hist1);
    scan1_k<<<1, 32, 0, stream>>>(hist1, nFg, sel1, rank2);
    hist2_k<<<grid, THREADS, 0, stream>>>(pbits, sel1, hist2);
    scan2_k<<<1, 32, 0, stream>>>(hist2, sel1, rank2, sel2, rank3);
    hist3_k<<<grid, THREADS, 0, stream>>>(pbits, sel1, sel2, hist3);
    scan3_k<<<1, 32, 0, stream>>>(hist3, sel1, sel2, rank3, thr);
    kept_k<<<grid, THREADS, 0, stream>>>(pbits, thr, keptSum, keptCnt);
    finalize_k<<<1, 32, 0, stream>>>(sumAll, sumFg, keptSum, keptCnt, out);
}